// IMUModel_19808389169186
// MI455X (gfx1250) — compile-verified
//
#include <hip/hip_runtime.h>
#include <hip/hip_bf16.h>

typedef __attribute__((ext_vector_type(2))) float v2f;
typedef __attribute__((ext_vector_type(8))) float v8f;

#define M_SEQ 1024
#define S_STEPS 256
#define K_KF 1025

__device__ __forceinline__ void quat_to_mat9(float qx, float qy, float qz, float qw,
                                             float R[3][3]) {
  R[0][0] = 1.f - 2.f * (qy * qy + qz * qz);
  R[0][1] = 2.f * (qx * qy - qz * qw);
  R[0][2] = 2.f * (qx * qz + qy * qw);
  R[1][0] = 2.f * (qx * qy + qz * qw);
  R[1][1] = 1.f - 2.f * (qx * qx + qz * qz);
  R[1][2] = 2.f * (qy * qz - qx * qw);
  R[2][0] = 2.f * (qx * qz - qy * qw);
  R[2][1] = 2.f * (qy * qz + qx * qw);
  R[2][2] = 1.f - 2.f * (qx * qx + qy * qy);
}

__device__ __forceinline__ float4 qmul4(float4 q, float4 r) {
  float4 o;
  o.x = q.w * r.x + q.x * r.w + q.y * r.z - q.z * r.y;
  o.y = q.w * r.y - q.x * r.z + q.y * r.w + q.z * r.x;
  o.z = q.w * r.z + q.x * r.y - q.y * r.x + q.z * r.w;
  o.w = q.w * r.w - q.x * r.x - q.y * r.y - q.z * r.z;
  return o;
}

// ---------------------------------------------------------------------------
// Phase A: one wave per sequence m. Serial scan over S steps; the 15x15
// Jacobian update J = (I + dt*F) J is padded to 16x16 and executed as a chain
// of four V_WMMA_F32_16X16X4_F32 ops (K split into 4 chunks of 4).
// F-hat and J round-trip through a per-workgroup 16x16 LDS tile each step.
// ---------------------------------------------------------------------------
__global__ __launch_bounds__(32) void imu_preint_kernel(
    const float* __restrict__ acc, const float* __restrict__ gyro,
    const float* __restrict__ dtv, const float* __restrict__ bias_a,
    const float* __restrict__ bias_w, float* __restrict__ ws) {
  __shared__ __align__(16) float sF[256];  // F-hat = I + dt*F (row-major 16x16)
  __shared__ __align__(16) float sJ[256];  // J (row-major 16x16, row/col 15 = 0 pad)

  const int m = blockIdx.x;
  const int l = threadIdx.x;  // 0..31 (single wave per workgroup)
  const int t = l >> 4;       // half-wave select
  const int col = l & 15;

  // Initialize F-hat (identity on rows 0..14; varying entries overwritten each
  // step) and J = I15 padded with zeros.
#pragma unroll
  for (int i = 0; i < 8; ++i) {
    int idx = l * 8 + i;
    int r = idx >> 4, c = idx & 15;
    float id = (r == c && r < 15) ? 1.f : 0.f;
    sF[idx] = id;
    sJ[idx] = id;
  }
  __syncthreads();

  const float* pa = acc + (size_t)m * S_STEPS * 3;
  const float* pg = gyro + (size_t)m * S_STEPS * 3;
  const float* pdt = dtv + (size_t)m * S_STEPS;
  const float* pba = bias_a + (size_t)m * S_STEPS * 3;
  const float* pbw = bias_w + (size_t)m * S_STEPS * 3;

  float qx = 0.f, qy = 0.f, qz = 0.f, qw = 1.f;
  float vx = 0.f, vy = 0.f, vz = 0.f;
  float px = 0.f, py = 0.f, pz = 0.f;
  float kfdt = 0.f;

  for (int s = 0; s < S_STEPS; ++s) {
    const float d = pdt[s];
    const float ax = pa[3 * s + 0], ay = pa[3 * s + 1], az = pa[3 * s + 2];
    const float gx = pg[3 * s + 0], gy = pg[3 * s + 1], gz = pg[3 * s + 2];
    const float bax = pba[3 * s + 0], bay = pba[3 * s + 1], baz = pba[3 * s + 2];
    const float bwx = pbw[3 * s + 0], bwy = pbw[3 * s + 1], bwz = pbw[3 * s + 2];
    kfdt += d;

    // a_w = R(q_old) @ a ; p,v updates
    float R[3][3];
    quat_to_mat9(qx, qy, qz, qw, R);
    float awx = R[0][0] * ax + R[0][1] * ay + R[0][2] * az;
    float awy = R[1][0] * ax + R[1][1] * ay + R[1][2] * az;
    float awz = R[2][0] * ax + R[2][1] * ay + R[2][2] * az;
    px += vx * d + 0.5f * awx * d * d;
    py += vy * d + 0.5f * awy * d * d;
    pz += vz * d + 0.5f * awz * d * d;
    vx += awx * d;
    vy += awy * d;
    vz += awz * d;

    // q = q * exp(w*dt)
    float phix = gx * d, phiy = gy * d, phiz = gz * d;
    float th2 = phix * phix + phiy * phiy + phiz * phiz;
    float th = sqrtf(th2);
    float half = 0.5f * th;
    float kk = (th < 1e-6f) ? (0.5f - th2 * (1.f / 48.f)) : (sinf(half) / th);
    float ex = phix * kk, ey = phiy * kk, ez = phiz * kk, ew = cosf(half);
    float nqx = qw * ex + qx * ew + qy * ez - qz * ey;
    float nqy = qw * ey - qx * ez + qy * ew + qz * ex;
    float nqz = qw * ez + qx * ey - qy * ex + qz * ew;
    float nqw = qw * ew - qx * ex - qy * ey - qz * ez;
    qx = nqx; qy = nqy; qz = nqz; qw = nqw;

    // F built from R(q_new), a-ba, w-bw
    float Rn[3][3];
    quat_to_mat9(qx, qy, qz, qw, Rn);
    float amx = ax - bax, amy = ay - bay, amz = az - baz;
    float wmx = gx - bwx, wmy = gy - bwy, wmz = gz - bwz;
    float RS[3][3];  // Rn @ skew(am)
#pragma unroll
    for (int i = 0; i < 3; ++i) {
      RS[i][0] = Rn[i][1] * amz - Rn[i][2] * amy;
      RS[i][1] = -Rn[i][0] * amz + Rn[i][2] * amx;
      RS[i][2] = Rn[i][0] * amy - Rn[i][1] * amx;
    }
    float SW[3][3] = {{0.f, -wmz, wmy}, {wmz, 0.f, -wmx}, {-wmy, wmx, 0.f}};

    // lane 0 scatters the 33 time-varying entries of F-hat (wave-uniform data)
    if (l == 0) {
      sF[0 * 16 + 3] = d;
      sF[1 * 16 + 4] = d;
      sF[2 * 16 + 5] = d;
#pragma unroll
      for (int i = 0; i < 3; ++i) {
#pragma unroll
        for (int j = 0; j < 3; ++j) {
          sF[(3 + i) * 16 + 6 + j] = -d * RS[i][j];           // -dt*R*skew(a-ba)
          sF[(3 + i) * 16 + 9 + j] = -d * Rn[i][j];           // -dt*R
          sF[(6 + i) * 16 + 6 + j] =
              ((i == j) ? 1.f : 0.f) - d * SW[i][j];          // I - dt*skew(w-bw)
        }
        sF[(6 + i) * 16 + 12 + i] = -d;                       // -dt*I
      }
    }
    __syncthreads();  // single-wave WG: S_NOP, acts as LDS ordering fence

    // A operand (F-hat), documented 16x4 f32 layout:
    //   vgpr j, lane l: A[row = l&15][k = 2*(l>>4) + j]
    v2f a0 = *(const v2f*)&sF[col * 16 + 0 + 2 * t];
    v2f a1 = *(const v2f*)&sF[col * 16 + 4 + 2 * t];
    v2f a2 = *(const v2f*)&sF[col * 16 + 8 + 2 * t];
    v2f a3 = *(const v2f*)&sF[col * 16 + 12 + 2 * t];
    // B operand (J), 4x16 layout: vgpr j, lane l: B[k = 2*(l>>4)+j][n = l&15]
    v2f b0, b1, b2, b3;
    b0.x = sJ[(0 + 2 * t) * 16 + col];
    b0.y = sJ[(1 + 2 * t) * 16 + col];
    b1.x = sJ[(4 + 2 * t) * 16 + col];
    b1.y = sJ[(5 + 2 * t) * 16 + col];
    b2.x = sJ[(8 + 2 * t) * 16 + col];
    b2.y = sJ[(9 + 2 * t) * 16 + col];
    b3.x = sJ[(12 + 2 * t) * 16 + col];
    b3.y = sJ[(13 + 2 * t) * 16 + col];

    v8f Dv = {};
    Dv = __builtin_amdgcn_wmma_f32_16x16x4_f32(false, a0, false, b0, (short)0, Dv, false, false);
    Dv = __builtin_amdgcn_wmma_f32_16x16x4_f32(false, a1, false, b1, (short)0, Dv, false, false);
    Dv = __builtin_amdgcn_wmma_f32_16x16x4_f32(false, a2, false, b2, (short)0, Dv, false, false);
    Dv = __builtin_amdgcn_wmma_f32_16x16x4_f32(false, a3, false, b3, (short)0, Dv, false, false);

    __syncthreads();
    // D layout: vgpr r, lane l -> J[r + 8*(l>>4)][l&15]
#pragma unroll
    for (int r = 0; r < 8; ++r) sJ[(r + 8 * t) * 16 + col] = Dv[r];
    __syncthreads();
  }

  if (l == 0) {
    float ba0[3] = {pba[0], pba[1], pba[2]};
    float bw0[3] = {pbw[0], pbw[1], pbw[2]};
    float ph[3] = {px, py, pz}, vh[3] = {vx, vy, vz};
    float al[3], be[3], dth[3];
#pragma unroll
    for (int i = 0; i < 3; ++i) {
      float s1 = 0.f, s2 = 0.f, s3 = 0.f, s4 = 0.f, s5 = 0.f;
#pragma unroll
      for (int j = 0; j < 3; ++j) {
        s1 += sJ[i * 16 + 9 + j] * ba0[j];
        s2 += sJ[i * 16 + 12 + j] * bw0[j];
        s3 += sJ[(3 + i) * 16 + 9 + j] * ba0[j];
        s4 += sJ[(3 + i) * 16 + 12 + j] * bw0[j];
        s5 += sJ[(6 + i) * 16 + 12 + j] * bw0[j];
      }
      al[i] = ph[i] + s1 + s2;
      be[i] = vh[i] + s3 + s4;
      dth[i] = 0.5f * s5;
    }
    // gamma = qmul(gamma_hat, [dtheta, 1])
    float gxo = qw * dth[0] + qx + qy * dth[2] - qz * dth[1];
    float gyo = qw * dth[1] - qx * dth[2] + qy + qz * dth[0];
    float gzo = qw * dth[2] + qx * dth[1] - qy * dth[0] + qz;
    float gwo = qw - qx * dth[0] - qy * dth[1] - qz * dth[2];
    ws[4 * m + 0] = gxo;
    ws[4 * m + 1] = gyo;
    ws[4 * m + 2] = gzo;
    ws[4 * m + 3] = gwo;
    ws[4 * M_SEQ + 3 * m + 0] = al[0];
    ws[4 * M_SEQ + 3 * m + 1] = al[1];
    ws[4 * M_SEQ + 3 * m + 2] = al[2];
    ws[7 * M_SEQ + 3 * m + 0] = be[0];
    ws[7 * M_SEQ + 3 * m + 1] = be[1];
    ws[7 * M_SEQ + 3 * m + 2] = be[2];
    ws[10 * M_SEQ + m] = kfdt;
  }
}

// ---------------------------------------------------------------------------
// Phase B: single 1024-thread workgroup. Quaternion product is associative ->
// Hillis-Steele scan (10 rounds) replaces the serial rot chain; two add-scans
// for vel/pos cumsums; then the error rows are written.
// ---------------------------------------------------------------------------
__global__ __launch_bounds__(1024) void imu_scan_kernel(
    const float* __restrict__ ws, const float* __restrict__ g,
    const float* __restrict__ init_rot, const float* __restrict__ init_pos,
    const float* __restrict__ init_vel, const float* __restrict__ gt_rot,
    const float* __restrict__ gt_pos, const float* __restrict__ gt_vel,
    float* __restrict__ out) {
  __shared__ float4 sQ[M_SEQ];
  __shared__ float4 sS[M_SEQ];
  const int i = threadIdx.x;

  sQ[i] = make_float4(ws[4 * i + 0], ws[4 * i + 1], ws[4 * i + 2], ws[4 * i + 3]);
  __syncthreads();
  // inclusive left-to-right quaternion product scan
  for (int off = 1; off < M_SEQ; off <<= 1) {
    float4 v = sQ[i];
    if (i >= off) v = qmul4(sQ[i - off], v);
    __syncthreads();
    sQ[i] = v;
    __syncthreads();
  }

  float4 q0 = make_float4(init_rot[0], init_rot[1], init_rot[2], init_rot[3]);
  float4 roti = (i == 0) ? q0 : qmul4(q0, sQ[i - 1]);  // rot[i]
  float4 rotK = qmul4(q0, sQ[M_SEQ - 1]);              // rot[1024]

  const float gx = g[0], gy = g[1], gz = g[2];
  const float kf = ws[10 * M_SEQ + i];
  const float alx = ws[4 * M_SEQ + 3 * i + 0];
  const float aly = ws[4 * M_SEQ + 3 * i + 1];
  const float alz = ws[4 * M_SEQ + 3 * i + 2];
  const float bex = ws[7 * M_SEQ + 3 * i + 0];
  const float bey = ws[7 * M_SEQ + 3 * i + 1];
  const float bez = ws[7 * M_SEQ + 3 * i + 2];

  float R[3][3];
  quat_to_mat9(roti.x, roti.y, roti.z, roti.w, R);
  float dvx = -gx * kf + R[0][0] * bex + R[0][1] * bey + R[0][2] * bez;
  float dvy = -gy * kf + R[1][0] * bex + R[1][1] * bey + R[1][2] * bez;
  float dvz = -gz * kf + R[2][0] * bex + R[2][1] * bey + R[2][2] * bez;

  sS[i] = make_float4(dvx, dvy, dvz, 0.f);
  __syncthreads();
  for (int off = 1; off < M_SEQ; off <<= 1) {
    float4 v = sS[i];
    if (i >= off) {
      float4 u = sS[i - off];
      v.x += u.x; v.y += u.y; v.z += u.z;
    }
    __syncthreads();
    sS[i] = v;
    __syncthreads();
  }
  float velx = init_vel[0] + (i ? sS[i - 1].x : 0.f);
  float vely = init_vel[1] + (i ? sS[i - 1].y : 0.f);
  float velz = init_vel[2] + (i ? sS[i - 1].z : 0.f);
  float velKx = init_vel[0] + sS[M_SEQ - 1].x;  // vel[1024]
  float velKy = init_vel[1] + sS[M_SEQ - 1].y;
  float velKz = init_vel[2] + sS[M_SEQ - 1].z;
  __syncthreads();  // all reads of sS done before reuse

  float dpx = velx * kf - 0.5f * gx * kf * kf + R[0][0] * alx + R[0][1] * aly + R[0][2] * alz;
  float dpy = vely * kf - 0.5f * gy * kf * kf + R[1][0] * alx + R[1][1] * aly + R[1][2] * alz;
  float dpz = velz * kf - 0.5f * gz * kf * kf + R[2][0] * alx + R[2][1] * aly + R[2][2] * alz;
  sS[i] = make_float4(dpx, dpy, dpz, 0.f);
  __syncthreads();
  for (int off = 1; off < M_SEQ; off <<= 1) {
    float4 v = sS[i];
    if (i >= off) {
      float4 u = sS[i - off];
      v.x += u.x; v.y += u.y; v.z += u.z;
    }
    __syncthreads();
    sS[i] = v;
    __syncthreads();
  }
  float posx = init_pos[0] + (i ? sS[i - 1].x : 0.f);
  float posy = init_pos[1] + (i ? sS[i - 1].y : 0.f);
  float posz = init_pos[2] + (i ? sS[i - 1].z : 0.f);
  float posKx = init_pos[0] + sS[M_SEQ - 1].x;  // pos[1024]
  float posKy = init_pos[1] + sS[M_SEQ - 1].y;
  float posKz = init_pos[2] + sS[M_SEQ - 1].z;

  auto emit = [&](int j, float4 rj, float vx_, float vy_, float vz_,
                  float px_, float py_, float pz_) {
    float4 gr = make_float4(gt_rot[4 * j + 0], gt_rot[4 * j + 1],
                            gt_rot[4 * j + 2], gt_rot[4 * j + 3]);
    float4 cr = make_float4(-gr.x, -gr.y, -gr.z, gr.w);
    float4 e = qmul4(cr, rj);
    float n = sqrtf(e.x * e.x + e.y * e.y + e.z * e.z);
    float theta = 2.f * atan2f(n, e.w);
    float scale = (n < 1e-6f) ? (2.f / ((fabsf(e.w) < 1e-6f) ? 1.f : e.w))
                              : (theta / n);
    out[3 * j + 0] = e.x * scale;
    out[3 * j + 1] = e.y * scale;
    out[3 * j + 2] = e.z * scale;
    float d0 = gt_pos[3 * j + 0] - px_;
    float d1 = gt_pos[3 * j + 1] - py_;
    float d2 = gt_pos[3 * j + 2] - pz_;
    out[3 * K_KF + 3 * j + 0] = d0 * d0;
    out[3 * K_KF + 3 * j + 1] = d1 * d1;
    out[3 * K_KF + 3 * j + 2] = d2 * d2;
    float e0 = gt_vel[3 * j + 0] - vx_;
    float e1 = gt_vel[3 * j + 1] - vy_;
    float e2 = gt_vel[3 * j + 2] - vz_;
    out[6 * K_KF + 3 * j + 0] = e0 * e0;
    out[6 * K_KF + 3 * j + 1] = e1 * e1;
    out[6 * K_KF + 3 * j + 2] = e2 * e2;
  };
  emit(i, roti, velx, vely, velz, posx, posy, posz);
  if (i == 0) emit(M_SEQ, rotK, velKx, velKy, velKz, posKx, posKy, posKz);
}

extern "C" void kernel_launch(void* const* d_in, const int* in_sizes, int n_in,
                              void* d_out, int out_size, void* d_ws, size_t ws_size,
                              hipStream_t stream) {
  (void)in_sizes; (void)n_in; (void)out_size; (void)ws_size;
  const float* acc = (const float*)d_in[0];
  const float* gyro = (const float*)d_in[1];
  const float* dt = (const float*)d_in[2];
  const float* bias_a = (const float*)d_in[3];
  const float* bias_w = (const float*)d_in[4];
  const float* g = (const float*)d_in[5];
  const float* init_rot = (const float*)d_in[6];
  const float* init_pos = (const float*)d_in[7];
  const float* init_vel = (const float*)d_in[8];
  const float* gt_rot = (const float*)d_in[9];
  const float* gt_pos = (const float*)d_in[10];
  const float* gt_vel = (const float*)d_in[11];
  float* ws = (float*)d_ws;
  float* out = (float*)d_out;

  imu_preint_kernel<<<dim3(M_SEQ), dim3(32), 0, stream>>>(acc, gyro, dt, bias_a,
                                                          bias_w, ws);
  imu_scan_kernel<<<dim3(1), dim3(M_SEQ), 0, stream>>>(
      ws, g, init_rot, init_pos, init_vel, gt_rot, gt_pos, gt_vel, out);
}